// UpsampleModule_18915035971606
// MI455X (gfx1250) — compile-verified
//
#include <hip/hip_runtime.h>
#include <hip/hip_bf16.h>
#include <math.h>

#define N_NODES 10000
#define M_PTS   20000
#define N_EDGES 160000
#define C_IN    128
#define F_OUT   128
#define K4C     (4*C_IN)   // 512
#define C3      (3*C_IN)   // 384

typedef __attribute__((ext_vector_type(2))) float v2f;
typedef __attribute__((ext_vector_type(8))) float v8f;

__device__ __forceinline__ float elu_f(float v) {
    return v > 0.f ? v : (expf(v) - 1.f);
}

// ---------------- stage 0: zero scratch (agg + cnt) ----------------
__global__ void zero_ws_kernel(float* __restrict__ p, int n) {
    int i = blockIdx.x * blockDim.x + threadIdx.x;
    if (i < n) p[i] = 0.f;
}

// ---------------- stage 1: edge scatter (DPOConv message + segment_sum) ----
__global__ void edge_scatter_kernel(const float* __restrict__ x,
                                    const float* __restrict__ pos,
                                    const int*   __restrict__ ei,
                                    float*       __restrict__ agg) {
    int idx = blockIdx.x * blockDim.x + threadIdx.x;   // E*C = 20.48M < 2^31
    if (idx >= N_EDGES * C_IN) return;
    int e = idx >> 7;          // C_IN == 128
    int c = idx & 127;
    int src = ei[e];
    int dst = ei[N_EDGES + e];
    float pix = pos[2*dst], piy = pos[2*dst + 1];
    float pjx = pos[2*src], pjy = pos[2*src + 1];
    float dxp = pix - pjx, dyp = piy - pjy;
    float r2 = dxp*dxp + dyp*dyp;
    float s  = 1.0f / (r2 + 0.01f);           // R_CORRECTION == 1
    float dx = dxp * s, dy = dyp * s;
    float vj = x[(long long)src * C_IN + c];
    float vi = x[(long long)dst * C_IN + c];
    float vd = vi - vj;
    float* base = agg + (long long)dst * C3;
    atomicAdd(base + c,           vd * dx);
    atomicAdd(base + C_IN + c,    vd * dy);
    atomicAdd(base + 2*C_IN + c,  vj);
}

__global__ void edge_count_kernel(const int* __restrict__ ei,
                                  float* __restrict__ cnt) {
    int e = blockIdx.x * blockDim.x + threadIdx.x;
    if (e >= N_EDGES) return;
    atomicAdd(&cnt[ei[N_EDGES + e]], 1.0f);
}

// ---------------- WMMA tile core (16 rows x 128 cols per block) ------------
// s_a: LDS A-tile [16][KDIM], 8 wave32 per block, wave w owns cols [16w,16w+16)
// Inner loop: ds_load_b64 (A frag) + 2x coalesced global_load_b32 (B frag)
// + v_wmma_f32_16x16x4_f32, unrolled x4.
template<int KDIM>
__device__ __forceinline__ void wmma_tile_compute(const float* __restrict__ s_a,
                                                  const float* __restrict__ W,
                                                  const float* __restrict__ bias,
                                                  float* __restrict__ out,
                                                  int tile_m) {
    const int tid  = threadIdx.x;
    const int wave = tid >> 5;       // 0..7 == tile_n
    const int lane = tid & 31;
    const int half = lane >> 4;
    const int mr   = lane & 15;
    const int nc   = lane & 15;
    const int n0   = wave * 16;
    const float* ar = s_a + mr * KDIM;
    const float* wp = W + n0 + nc;

    v8f c = {};
    #pragma unroll 4
    for (int k0 = 0; k0 < KDIM; k0 += 4) {
        int ka = k0 + 2 * half;                  // even -> 8B aligned
        v2f a = *(const v2f*)(ar + ka);
        v2f b;
        b.x = wp[ka * F_OUT];
        b.y = wp[(ka + 1) * F_OUT];
        c = __builtin_amdgcn_wmma_f32_16x16x4_f32(
                false, a, false, b, (short)0, c, false, false);
    }
    float bv = bias[n0 + nc];
    #pragma unroll
    for (int v = 0; v < 8; ++v) {
        int m = tile_m * 16 + v + 8 * half;
        out[(long long)m * F_OUT + n0 + nc] = elu_f(c[v] + bv);
    }
}

// ---------------- stage 2: GEMM1  h1 = ELU([agg/cnt, x] @ W_conv + b) ------
__global__ void __launch_bounds__(256)
gemm1_wmma_kernel(const float* __restrict__ agg,
                  const float* __restrict__ cnt,
                  const float* __restrict__ x,
                  const float* __restrict__ W,     // [512,128]
                  const float* __restrict__ bias,  // [128]
                  float* __restrict__ out) {       // [N,128]
    __shared__ __align__(16) float s_a[16 * K4C];  // 32 KB
    __shared__ float s_inv[16];
    const int tile_m = blockIdx.x;                 // 0..624
    const int tid = threadIdx.x;
    if (tid < 16) s_inv[tid] = 1.0f / fmaxf(cnt[tile_m * 16 + tid], 1.0f);
    __syncthreads();
    // stage A tile: [agg/cnt (384) | x (128)] per row, predicate resolved here
    for (int i = tid; i < 16 * K4C; i += 256) {
        int r    = i >> 9;         // / 512
        int ccol = i & 511;
        int row  = tile_m * 16 + r;
        s_a[i] = (ccol < C3)
               ? agg[(long long)row * C3 + ccol] * s_inv[r]
               : x[(long long)row * C_IN + (ccol - C3)];
    }
    __syncthreads();
    wmma_tile_compute<K4C>(s_a, W, bias, out, tile_m);
}

// ---------------- stage 3: GEMM2  h2 = ELU(h1 @ W2 + b2) -------------------
__global__ void __launch_bounds__(256)
gemm2_wmma_kernel(const float* __restrict__ A,     // [N,128]
                  const float* __restrict__ W,     // [128,128]
                  const float* __restrict__ bias,  // [128]
                  float* __restrict__ out) {       // [N,128]
    __shared__ __align__(16) float s_a[16 * F_OUT];  // 8 KB
    const int tile_m = blockIdx.x;
    const int tid = threadIdx.x;
    for (int i = tid; i < 16 * F_OUT; i += 256) {
        int r    = i >> 7;
        int ccol = i & 127;
        s_a[i] = A[(long long)(tile_m * 16 + r) * F_OUT + ccol];
    }
    __syncthreads();
    wmma_tile_compute<F_OUT>(s_a, W, bias, out, tile_m);
}

// ---------------- stage 4: brute-force 3-NN with LDS-tiled source points ---
#define KNN_CHUNK 1024
__global__ void knn_kernel(const float* __restrict__ pos,       // [N,2] sources
                           const float* __restrict__ pos_skip,  // [M,2] targets
                           int*   __restrict__ knn_idx,         // [M,3]
                           float* __restrict__ knn_w) {         // [M,3]
    __shared__ float2 sp[KNN_CHUNK];
    int m = blockIdx.x * blockDim.x + threadIdx.x;
    float tx = 0.f, ty = 0.f;
    if (m < M_PTS) { tx = pos_skip[2*m]; ty = pos_skip[2*m + 1]; }
    float d0 = INFINITY, d1 = INFINITY, d2 = INFINITY;
    int   i0 = 0, i1 = 0, i2 = 0;
    for (int base = 0; base < N_NODES; base += KNN_CHUNK) {
        int cn = min(KNN_CHUNK, N_NODES - base);
        for (int t = threadIdx.x; t < cn; t += blockDim.x)
            sp[t] = make_float2(pos[2*(base + t)], pos[2*(base + t) + 1]);
        __syncthreads();
        for (int t = 0; t < cn; ++t) {
            float ddx = tx - sp[t].x, ddy = ty - sp[t].y;
            float d = ddx*ddx + ddy*ddy;
            int gi = base + t;
            if (d < d2) {
                if (d < d0)      { d2=d1; i2=i1; d1=d0; i1=i0; d0=d; i0=gi; }
                else if (d < d1) { d2=d1; i2=i1; d1=d;  i1=gi; }
                else             { d2=d;  i2=gi; }
            }
        }
        __syncthreads();
    }
    if (m < M_PTS) {
        knn_idx[3*m]     = i0;
        knn_idx[3*m + 1] = i1;
        knn_idx[3*m + 2] = i2;
        knn_w[3*m]       = 1.0f / fmaxf(d0, 1e-16f);
        knn_w[3*m + 1]   = 1.0f / fmaxf(d1, 1e-16f);
        knn_w[3*m + 2]   = 1.0f / fmaxf(d2, 1e-16f);
    }
}

// ---------------- stage 5: IDW gather to output ----------------------------
__global__ void interp_kernel(const float* __restrict__ h,        // [N,128]
                              const int*   __restrict__ knn_idx,
                              const float* __restrict__ knn_w,
                              float* __restrict__ out) {          // [M,128]
    int idx = blockIdx.x * blockDim.x + threadIdx.x;  // M*F = 2.56M
    if (idx >= M_PTS * F_OUT) return;
    int m = idx >> 7;
    int f = idx & 127;
    int i0 = knn_idx[3*m], i1 = knn_idx[3*m + 1], i2 = knn_idx[3*m + 2];
    float w0 = knn_w[3*m], w1 = knn_w[3*m + 1], w2 = knn_w[3*m + 2];
    float s = w0 * h[(long long)i0 * F_OUT + f]
            + w1 * h[(long long)i1 * F_OUT + f]
            + w2 * h[(long long)i2 * F_OUT + f];
    out[idx] = s / (w0 + w1 + w2);
}

// ---------------- host launcher -------------------------------------------
extern "C" void kernel_launch(void* const* d_in, const int* in_sizes, int n_in,
                              void* d_out, int out_size, void* d_ws, size_t ws_size,
                              hipStream_t stream) {
    const float* x        = (const float*)d_in[0];   // [N,128]
    const float* pos      = (const float*)d_in[1];   // [N,2]
    const float* pos_skip = (const float*)d_in[2];   // [M,2]
    const int*   ei       = (const int*)d_in[3];     // [2,E]
    const float* W_conv   = (const float*)d_in[4];   // [512,128]
    const float* b_conv   = (const float*)d_in[5];   // [128]
    const float* W2       = (const float*)d_in[6];   // [128,128]
    const float* b2       = (const float*)d_in[7];   // [128]
    float* out = (float*)d_out;                      // [M,128]

    // workspace layout (floats): agg | cnt | h1 | h2 | knn_w | knn_idx(int)
    float* agg   = (float*)d_ws;
    float* cnt   = agg + (size_t)N_NODES * C3;
    float* h1    = cnt + N_NODES;
    float* h2    = h1 + (size_t)N_NODES * F_OUT;
    float* knn_w = h2 + (size_t)N_NODES * F_OUT;
    int*   knn_i = (int*)(knn_w + (size_t)M_PTS * 3);

    // stage 0: zero agg + cnt (contiguous)
    int zn = N_NODES * C3 + N_NODES;
    zero_ws_kernel<<<(zn + 255) / 256, 256, 0, stream>>>(agg, zn);

    // stage 1: scatter messages + degree counts
    edge_scatter_kernel<<<(N_EDGES * C_IN + 255) / 256, 256, 0, stream>>>(x, pos, ei, agg);
    edge_count_kernel<<<(N_EDGES + 255) / 256, 256, 0, stream>>>(ei, cnt);

    // stage 2/3: WMMA GEMMs (one 16-row tile per block, 8 wave32 tiles wide)
    gemm1_wmma_kernel<<<N_NODES / 16, 256, 0, stream>>>(agg, cnt, x, W_conv, b_conv, h1);
    gemm2_wmma_kernel<<<N_NODES / 16, 256, 0, stream>>>(h1, W2, b2, h2);

    // stage 4/5: kNN(3) + inverse-distance interpolation
    knn_kernel<<<(M_PTS + 255) / 256, 256, 0, stream>>>(pos, pos_skip, knn_i, knn_w);
    interp_kernel<<<(M_PTS * F_OUT + 255) / 256, 256, 0, stream>>>(h2, knn_i, knn_w, out);
}